// RSSM_32701880991915
// MI455X (gfx1250) — compile-verified
//
#include <hip/hip_runtime.h>
#include <math.h>

typedef __attribute__((ext_vector_type(2))) float v2f;
typedef __attribute__((ext_vector_type(8))) float v8f;

#define B_DIM 1024
#define T_DIM 64
#define E_DIM 1024
#define A_DIM 10
#define S_DIM 30
#define H_DIM 200
#define OUT_STRIDE 580

// workspace layout (float offsets)
#define EPROJ_OFF   0L
#define EPROJ_SZ    ((long)T_DIM * B_DIM * H_DIM)          // 13107200
#define P_IMG1_OFF  (EPROJ_OFF + EPROJ_SZ)                 // 208*40
#define P_WI_OFF    (P_IMG1_OFF + 208L * 40)               // 608*200
#define P_WH_OFF    (P_WI_OFF + 608L * 200)                // 608*200
#define P_IMG2_OFF  (P_WH_OFF + 608L * 200)                // 208*200
#define P_IMG3_OFF  (P_IMG2_OFF + 208L * 200)              // 64*200
#define P_OBS1D_OFF (P_IMG3_OFF + 64L * 200)               // 208*200
#define P_OBS2_OFF  (P_OBS1D_OFF + 208L * 200)             // 64*200
#define P_OBS1E_OFF (P_OBS2_OFF + 64L * 200)               // 208*1024

__device__ __forceinline__ float sigmoid_f(float x) { return 1.0f / (1.0f + __expf(-x)); }
__device__ __forceinline__ float elu_f(float x)     { return x > 0.0f ? x : __expf(x) - 1.0f; }
__device__ __forceinline__ float softplus_f(float x){ return x > 20.0f ? x : log1pf(__expf(x)); }

// ---------------------------------------------------------------------------
// Weight repack: dst[col][k] = (col < N) ? src[k*ldsrc + col] : 0
// dst is [Npad][K], transposed + zero-padded so WMMA B loads are
// unconditional contiguous b64 per lane.
// ---------------------------------------------------------------------------
__global__ void __launch_bounds__(256)
pack_wt_kernel(const float* __restrict__ src, float* __restrict__ dst,
               int K, int N, int Npad, int ldsrc)
{
  const long i = (long)blockIdx.x * 256 + threadIdx.x;
  const long total = (long)Npad * K;
  if (i >= total) return;
  const int col = (int)(i / K);
  const int k   = (int)(i - (long)col * K);
  dst[i] = (col < N) ? src[(long)k * ldsrc + col] : 0.0f;
}

// acc += A(16 x K, LDS row-major, lda) * Wt-column-block, where Wt points at
// the packed transposed column for THIS lane: Wt = pack + col*Kstride.
// No guards: padded columns are zero.  One ds_load_b64 + one global_load_b64
// per v_wmma_f32_16x16x4_f32.
__device__ __forceinline__ void wmma_gemm_t(v8f& acc, const float* sA, int lda,
                                            const float* __restrict__ Wt, int K)
{
  const int lane = threadIdx.x & 31;
  const int l16  = lane & 15;
  const int koff = (lane >> 4) << 1;   // 0 or 2
  const float* a0 = sA + l16 * lda + koff;
  const float* w0 = Wt + koff;
#pragma unroll 4
  for (int k = 0; k < K; k += 4) {
    const v2f a = *(const v2f*)(a0 + k);
    const v2f b = *(const v2f*)(w0 + k);
    acc = __builtin_amdgcn_wmma_f32_16x16x4_f32(false, a, false, b, (short)0, acc,
                                                false, false);
  }
}

// ---------------------------------------------------------------------------
// Kernel A: eproj[t][b][h] = embed[b,t,:] @ obs1_w[H:,:] + obs1_b[h]
// Fully parallel precompute of the K=1024 GEMM (the FLOP majority), removing
// it from the serial scan.  4096 blocks x 8 waves, K staged via LDS.
// ---------------------------------------------------------------------------
__global__ void __launch_bounds__(256)
eproj_kernel(const float* __restrict__ embed,
             const float* __restrict__ p_obs1e,   // [208][1024] transposed
             const float* __restrict__ obs1_b,
             float* __restrict__ eproj)
{
  __shared__ float sA[16 * 260];
  const int mt = blockIdx.x;        // 0..4095
  const int t  = mt >> 6;
  const int b0 = (mt & 63) << 4;
  const int tid  = threadIdx.x;
  const int w    = tid >> 5;
  const int lane = tid & 31;
  const int l16  = lane & 15;
  const int hh   = lane >> 4;

  v8f acc0 = {};
  v8f acc1 = {};
  const int nt1 = w + 8;
  const int c0  = w * 16 + l16;          // < 208 always
  const int c1  = nt1 * 16 + l16;

  for (int ch = 0; ch < 4; ++ch) {
    __syncthreads();
    for (int i = tid; i < 16 * 64; i += 256) {   // 16 rows x 256 floats as float4
      const int m = i >> 6;
      const int q = (i & 63) << 2;
      const float4 v = *(const float4*)(embed +
          (((long)(b0 + m) * T_DIM + t) * E_DIM + ch * 256 + q));
      *(float4*)(&sA[m * 260 + q]) = v;
    }
    __syncthreads();
    wmma_gemm_t(acc0, sA, 260, p_obs1e + (long)c0 * E_DIM + ch * 256, 256);
    if (nt1 < 13)
      wmma_gemm_t(acc1, sA, 260, p_obs1e + (long)c1 * E_DIM + ch * 256, 256);
  }

  if (c0 < H_DIM) {
    const float bias = obs1_b[c0];
#pragma unroll
    for (int r = 0; r < 8; ++r)
      eproj[((long)t * B_DIM + (b0 + r + hh * 8)) * H_DIM + c0] = acc0[r] + bias;
  }
  if (nt1 < 13 && c1 < H_DIM) {
    const float bias = obs1_b[c1];
#pragma unroll
    for (int r = 0; r < 8; ++r)
      eproj[((long)t * B_DIM + (b0 + r + hh * 8)) * H_DIM + c1] = acc1[r] + bias;
  }
}

// ---------------------------------------------------------------------------
// Kernel B: the sequential scan. Each workgroup owns 16 batch rows; carried
// (stoch, deter) live in LDS across the T=64 loop.  64 blocks x 8 waves.
// ---------------------------------------------------------------------------
__global__ void __launch_bounds__(256)
rssm_kernel(const float* __restrict__ action,
            const float* __restrict__ eps_post,
            const float* __restrict__ eps_prior,
            const float* __restrict__ p_img1, const float* __restrict__ img1_b,
            const float* __restrict__ p_wi,   const float* __restrict__ p_wh,
            const float* __restrict__ gru_bi, const float* __restrict__ gru_bh,
            const float* __restrict__ p_img2, const float* __restrict__ img2_b,
            const float* __restrict__ p_img3, const float* __restrict__ img3_b,
            const float* __restrict__ p_obs1d,
            const float* __restrict__ p_obs2, const float* __restrict__ obs2_b,
            const float* __restrict__ eproj,
            float* __restrict__ out)
{
  __shared__ float s_deter[16 * 208];  // carried deter
  __shared__ float s_in1[16 * 48];     // [stoch(30) | action(10) | pad]
  __shared__ float bufA[16 * 208];     // x (img1 out)  /  o1 (obs1 out)
  __shared__ float bufG[16 * 416];     // r,z preacts + d_new ; later h2
  __shared__ float s_y[16 * 64];       // prior head raw params
  __shared__ float s_o2[16 * 64];      // posterior head raw params

  const int b0   = blockIdx.x << 4;
  const int tid  = threadIdx.x;
  const int w    = tid >> 5;
  const int lane = tid & 31;
  const int l16  = lane & 15;
  const int hh   = lane >> 4;

  for (int i = tid; i < 16 * 208; i += 256) s_deter[i] = 0.0f;
  for (int i = tid; i < 16 * 48;  i += 256) s_in1[i]   = 0.0f;
  __syncthreads();

  for (int t = 0; t < T_DIM; ++t) {
    // ---- stage 0: action slice into s_in1 cols 30..39 ----
    for (int i = tid; i < 16 * A_DIM; i += 256) {
      const int m = i / A_DIM;
      const int k = i - m * A_DIM;
      s_in1[m * 48 + S_DIM + k] = action[((long)(b0 + m) * T_DIM + t) * A_DIM + k];
    }
    __syncthreads();

    // ---- stage 1: x = elu([stoch,a] @ img1_w + b) -> bufA ----
    for (int nt = w; nt < 13; nt += 8) {
      const int col = nt * 16 + l16;        // < 208
      v8f acc = {};
      wmma_gemm_t(acc, s_in1, 48, p_img1 + (long)col * 40, S_DIM + A_DIM);
      if (col < H_DIM) {
        const float bias = img1_b[col];
#pragma unroll
        for (int r = 0; r < 8; ++r)
          bufA[(r + hh * 8) * 208 + col] = elu_f(acc[r] + bias);
      }
    }
    __syncthreads();

    // ---- stage 2a: r,z pre-activations (cols 0..399) -> bufG ----
    for (int nt = w; nt < 25; nt += 8) {
      const int col = nt * 16 + l16;        // < 400
      v8f acc = {};
      wmma_gemm_t(acc, bufA,    208, p_wi + (long)col * H_DIM, H_DIM);
      wmma_gemm_t(acc, s_deter, 208, p_wh + (long)col * H_DIM, H_DIM);
      const float bias = gru_bi[col] + gru_bh[col];
#pragma unroll
      for (int r = 0; r < 8; ++r)
        bufG[(r + hh * 8) * 416 + col] = acc[r] + bias;
    }
    __syncthreads();

    // ---- stage 2b: n-section split (i_n, h_n in regs) + GRU update.
    // d_new goes into bufG's dead r-slot: avoids WAR on s_deter, which other
    // waves still stream as the WMMA A operand. ----
    for (int nt = w; nt < 13; nt += 8) {
      const int h   = nt * 16 + l16;        // < 208
      const int col = 2 * H_DIM + h;        // < 608
      v8f acc_i = {};
      v8f acc_h = {};
      wmma_gemm_t(acc_i, bufA,    208, p_wi + (long)col * H_DIM, H_DIM);
      wmma_gemm_t(acc_h, s_deter, 208, p_wh + (long)col * H_DIM, H_DIM);
      if (h < H_DIM) {
        const float bi_n = gru_bi[col];
        const float bh_n = gru_bh[col];
#pragma unroll
        for (int r = 0; r < 8; ++r) {
          const int m = r + hh * 8;
          const float rr = sigmoid_f(bufG[m * 416 + h]);
          const float zz = sigmoid_f(bufG[m * 416 + H_DIM + h]);
          const float nn = tanhf(acc_i[r] + bi_n + rr * (acc_h[r] + bh_n));
          bufG[m * 416 + h] = (1.0f - zz) * nn + zz * s_deter[m * 208 + h];
        }
      }
    }
    __syncthreads();

    // ---- stage 3: commit deter, write both deter output slots ----
    for (int i = tid; i < 16 * H_DIM; i += 256) {
      const int m = i / H_DIM;
      const int h = i - m * H_DIM;
      const float d = bufG[m * 416 + h];
      s_deter[m * 208 + h] = d;
      float* orow = out + ((long)(b0 + m) * T_DIM + t) * OUT_STRIDE;
      orow[90 + h]  = d;
      orow[380 + h] = d;
    }
    __syncthreads();

    // ---- stage 4: h2 = elu(deter @ img2_w + b) -> bufG cols 0..199 ----
    for (int nt = w; nt < 13; nt += 8) {
      const int col = nt * 16 + l16;
      v8f acc = {};
      wmma_gemm_t(acc, s_deter, 208, p_img2 + (long)col * H_DIM, H_DIM);
      if (col < H_DIM) {
        const float bias = img2_b[col];
#pragma unroll
        for (int r = 0; r < 8; ++r)
          bufG[(r + hh * 8) * 416 + col] = elu_f(acc[r] + bias);
      }
    }
    __syncthreads();

    // ---- stage 5+6 fused: img3 (4 tiles) and obs1 deter-part (13 tiles) ----
    for (int j = w; j < 17; j += 8) {
      if (j < 4) {
        const int col = j * 16 + l16;       // < 64
        v8f acc = {};
        wmma_gemm_t(acc, bufG, 416, p_img3 + (long)col * H_DIM, H_DIM);
        if (col < 2 * S_DIM) {
          const float bias = img3_b[col];
#pragma unroll
          for (int r = 0; r < 8; ++r)
            s_y[(r + hh * 8) * 64 + col] = acc[r] + bias;
        }
      } else {
        const int col = (j - 4) * 16 + l16; // < 208
        v8f acc = {};
        wmma_gemm_t(acc, s_deter, 208, p_obs1d + (long)col * H_DIM, H_DIM);
        if (col < H_DIM) {
#pragma unroll
          for (int r = 0; r < 8; ++r) {
            const int m = r + hh * 8;
            const float ep = eproj[((long)t * B_DIM + (b0 + m)) * H_DIM + col];
            bufA[m * 208 + col] = elu_f(acc[r] + ep);   // eproj already has bias
          }
        }
      }
    }
    __syncthreads();

    // ---- stage 7: o2 = o1 @ obs2_w + b (waves 0..3) ----
    for (int j = w; j < 4; j += 8) {
      const int col = j * 16 + l16;         // < 64
      v8f acc = {};
      wmma_gemm_t(acc, bufA, 208, p_obs2 + (long)col * H_DIM, H_DIM);
      if (col < 2 * S_DIM) {
        const float bias = obs2_b[col];
#pragma unroll
        for (int r = 0; r < 8; ++r)
          s_o2[(r + hh * 8) * 64 + col] = acc[r] + bias;
      }
    }
    __syncthreads();

    // ---- stage 8: softplus heads, rsample, outputs, carry stoch ----
    for (int i = tid; i < 16 * S_DIM; i += 256) {
      const int m = i / S_DIM;
      const int s = i - m * S_DIM;
      const long bb = b0 + m;
      float* orow = out + (bb * T_DIM + t) * OUT_STRIDE;
      const long eidx = ((long)t * B_DIM + bb) * S_DIM + s;
      const float pm   = s_y[m * 64 + s];
      const float pstd = softplus_f(s_y[m * 64 + S_DIM + s]) + 0.1f;
      orow[290 + s] = pm;
      orow[320 + s] = pstd;
      orow[350 + s] = pm + pstd * eps_prior[eidx];
      const float qm   = s_o2[m * 64 + s];
      const float qstd = softplus_f(s_o2[m * 64 + S_DIM + s]) + 0.1f;
      const float qst  = qm + qstd * eps_post[eidx];
      orow[0 + s]  = qm;
      orow[30 + s] = qstd;
      orow[60 + s] = qst;
      s_in1[m * 48 + s] = qst;   // carried stoch
    }
    __syncthreads();
  }
}

static inline void launch_pack(const float* src, float* dst, int K, int N, int Npad,
                               int ldsrc, hipStream_t stream)
{
  const long total = (long)Npad * K;
  const int blocks = (int)((total + 255) / 256);
  pack_wt_kernel<<<dim3(blocks), dim3(256), 0, stream>>>(src, dst, K, N, Npad, ldsrc);
}

extern "C" void kernel_launch(void* const* d_in, const int* in_sizes, int n_in,
                              void* d_out, int out_size, void* d_ws, size_t ws_size,
                              hipStream_t stream)
{
  (void)in_sizes; (void)n_in; (void)out_size; (void)ws_size;
  const float* embed     = (const float*)d_in[0];
  const float* action    = (const float*)d_in[1];
  const float* eps_post  = (const float*)d_in[2];
  const float* eps_prior = (const float*)d_in[3];
  const float* img1_w = (const float*)d_in[4];
  const float* img1_b = (const float*)d_in[5];
  const float* gru_wi = (const float*)d_in[6];
  const float* gru_wh = (const float*)d_in[7];
  const float* gru_bi = (const float*)d_in[8];
  const float* gru_bh = (const float*)d_in[9];
  const float* img2_w = (const float*)d_in[10];
  const float* img2_b = (const float*)d_in[11];
  const float* img3_w = (const float*)d_in[12];
  const float* img3_b = (const float*)d_in[13];
  const float* obs1_w = (const float*)d_in[14];
  const float* obs1_b = (const float*)d_in[15];
  const float* obs2_w = (const float*)d_in[16];
  const float* obs2_b = (const float*)d_in[17];
  float* out = (float*)d_out;
  float* ws  = (float*)d_ws;

  float* eproj   = ws + EPROJ_OFF;
  float* p_img1  = ws + P_IMG1_OFF;
  float* p_wi    = ws + P_WI_OFF;
  float* p_wh    = ws + P_WH_OFF;
  float* p_img2  = ws + P_IMG2_OFF;
  float* p_img3  = ws + P_IMG3_OFF;
  float* p_obs1d = ws + P_OBS1D_OFF;
  float* p_obs2  = ws + P_OBS2_OFF;
  float* p_obs1e = ws + P_OBS1E_OFF;

  // one-time (per call) weight repack: transpose + pad columns to x16
  launch_pack(img1_w,              p_img1,  40,   H_DIM,     208, H_DIM,     stream);
  launch_pack(gru_wi,              p_wi,    H_DIM, 3 * H_DIM, 608, 3 * H_DIM, stream);
  launch_pack(gru_wh,              p_wh,    H_DIM, 3 * H_DIM, 608, 3 * H_DIM, stream);
  launch_pack(img2_w,              p_img2,  H_DIM, H_DIM,     208, H_DIM,     stream);
  launch_pack(img3_w,              p_img3,  H_DIM, 2 * S_DIM, 64,  2 * S_DIM, stream);
  launch_pack(obs1_w,              p_obs1d, H_DIM, H_DIM,     208, H_DIM,     stream);
  launch_pack(obs2_w,              p_obs2,  H_DIM, 2 * S_DIM, 64,  2 * S_DIM, stream);
  launch_pack(obs1_w + (long)H_DIM * H_DIM, p_obs1e, E_DIM, H_DIM, 208, H_DIM, stream);

  eproj_kernel<<<dim3(T_DIM * (B_DIM / 16)), dim3(256), 0, stream>>>(
      embed, p_obs1e, obs1_b, eproj);
  rssm_kernel<<<dim3(B_DIM / 16), dim3(256), 0, stream>>>(
      action, eps_post, eps_prior, p_img1, img1_b, p_wi, p_wh, gru_bi, gru_bh,
      p_img2, img2_b, p_img3, img3_b, p_obs1d, p_obs2, obs2_b, eproj, out);
}